// WeightedSum_10471130268471
// MI455X (gfx1250) — compile-verified
//
#include <hip/hip_runtime.h>

typedef __attribute__((ext_vector_type(2))) float        v2f;
typedef __attribute__((ext_vector_type(8))) float        v8f;
typedef __attribute__((ext_vector_type(4))) unsigned int v4u;
typedef __attribute__((ext_vector_type(4))) int          v4i;
typedef __attribute__((ext_vector_type(8))) int          v8i;

constexpr int kDim          = 128;                 // feature dim D
constexpr int kWavesPerBlk  = 8;
constexpr int kTileRows     = 16;                  // WMMA M
constexpr int kTilesPerWave = 8;
constexpr int kRowsPerWave  = kTileRows * kTilesPerWave;   // 128
constexpr int kTileFloats   = kTileRows * kDim;            // 2048 (8KB)

// ---- TDM: one DMA descriptor moves a rows x 128 f32 tile (row-major,
// stride 128) from global into LDS. Tracked with TENSORcnt (per-wave). ----
__device__ __forceinline__ void tdm_load_tile(const float* gsrc, float* ldst,
                                              int rows, unsigned rowsAvail) {
  v4u g0; v8i g1;
  v4i g2 = {0, 0, 0, 0};
  v4i g3 = {0, 0, 0, 0};
  v8i g4 = {0, 0, 0, 0, 0, 0, 0, 0};          // unused trailing group (VADDR4 slot)
  const unsigned ldsOff = (unsigned)(unsigned long long)(uintptr_t)ldst; // LDS byte addr
  const unsigned long long ga = (unsigned long long)(uintptr_t)gsrc;     // global byte addr
  // group0: count=1 | lds_addr | global_addr[56:0] | type=2
  g0[0] = 1u;
  g0[1] = ldsOff;
  g0[2] = (unsigned)ga;
  g0[3] = (unsigned)(ga >> 32) | (2u << 30);
  // group1: data_size=4B, tensor_dim0=128, tensor_dim1=rowsAvail,
  //         tile_dim0=128, tile_dim1=rows, tensor_dim0_stride=128
  const unsigned d1 = rowsAvail;
  g1[0] = (int)0x00020000u;                                  // data_size=2 (4 bytes)
  g1[1] = (int)(128u << 16);                                 // tensor_dim0[15:0] @ bits 63:48
  g1[2] = (int)((d1 & 0xFFFFu) << 16);                       // dim0 hi=0 | tensor_dim1[15:0]
  g1[3] = (int)(((d1 >> 16) & 0xFFFFu) | (128u << 16));      // tensor_dim1 hi | tile_dim0=128
  g1[4] = (int)(unsigned)rows;                               // tile_dim1=rows, tile_dim2=0
  g1[5] = 128;                                               // tensor_dim0_stride lo
  g1[6] = 0;
  g1[7] = 0;
  __builtin_amdgcn_tensor_load_to_lds(g0, g1, g2, g3, g4, 0);
}

__device__ __forceinline__ float sigmoidf_dev(float z) {
  return 1.0f / (1.0f + __expf(-z));
}

__global__ __launch_bounds__(256) void wsum_kernel(
    const float* __restrict__ x, const int* __restrict__ batch,
    const float* __restrict__ mask, const float* __restrict__ W,
    const float* __restrict__ bvec, float* __restrict__ out, int N) {
  __shared__ __align__(16) float ldsTiles[kWavesPerBlk * 2 * kTileFloats]; // 128 KB
  __shared__ __align__(16) float Wlds[kDim];

  if (threadIdx.x < kDim) Wlds[threadIdx.x] = W[threadIdx.x];
  __syncthreads();

  const int wid     = threadIdx.x >> 5;
  const int lane    = threadIdx.x & 31;
  const int waveIdx = blockIdx.x * kWavesPerBlk + wid;
  const int rowBase = waveIdx * kRowsPerWave;
  if (rowBase >= N) return;

  const float bias    = bvec[0];
  int totalRows       = N - rowBase;
  if (totalRows > kRowsPerWave) totalRows = kRowsPerWave;
  const int nTiles    = (totalRows + kTileRows - 1) / kTileRows;

  float* buf0 = &ldsTiles[(wid * 2 + 0) * kTileFloats];
  float* buf1 = &ldsTiles[(wid * 2 + 1) * kTileFloats];

  // prime the double-buffered TDM pipeline
  {
    int r0 = totalRows < kTileRows ? totalRows : kTileRows;
    tdm_load_tile(x + (size_t)rowBase * kDim, buf0, r0, (unsigned)(N - rowBase));
  }

  int   curSeg = batch[rowBase];
  float ax = 0.f, ay = 0.f, az = 0.f, aw = 0.f;

  const int half = (lane >> 4) << 1;   // K sub-offset: 0 (lanes 0-15) / 2 (lanes 16-31)
  const int mrow = lane & 15;          // A-matrix row M for this lane

  for (int t = 0; t < nTiles; ++t) {
    const int tb   = rowBase + t * kTileRows;
    int rows       = totalRows - t * kTileRows;
    if (rows > kTileRows) rows = kTileRows;

    if (t + 1 < nTiles) {
      const int ntb = tb + kTileRows;
      int rowsN = totalRows - (t + 1) * kTileRows;
      if (rowsN > kTileRows) rowsN = kTileRows;
      tdm_load_tile(x + (size_t)ntb * kDim, ((t + 1) & 1) ? buf1 : buf0, rowsN,
                    (unsigned)(N - ntb));
      __builtin_amdgcn_s_wait_tensorcnt(1);   // tile t complete (TDM in-order)
    } else {
      __builtin_amdgcn_s_wait_tensorcnt(0);
    }
    asm volatile("" ::: "memory");

    const float* tile = (t & 1) ? buf1 : buf0;

    // ---- gates: chain 32x V_WMMA_F32_16X16X4_F32, W broadcast over N ----
    v8f c = {0.f, 0.f, 0.f, 0.f, 0.f, 0.f, 0.f, 0.f};
#pragma unroll
    for (int k0 = 0; k0 < kDim; k0 += 4) {
      v2f a = *(const v2f*)(tile + mrow * kDim + k0 + half);  // A: x[row][k0+half..+1]
      v2f b = *(const v2f*)(Wlds + k0 + half);                // B: W[k0+half..+1] (all N cols)
      c = __builtin_amdgcn_wmma_f32_16x16x4_f32(false, a, false, b, (short)0, c,
                                                false, false);
    }
    // lane<16 holds rows v (v=0..7); lane>=16 holds rows v+8
    float g[8];
#pragma unroll
    for (int v = 0; v < 8; ++v) g[v] = sigmoidf_dev(c[v] + bias);

    // ---- gating multiply + sorted-run segmented accumulation ----
#pragma unroll
    for (int r = 0; r < kTileRows; ++r) {
      if (r < rows) {
        const int   gr  = tb + r;
        const int   seg = batch[gr];
        const float mk  = mask[gr];
        const float gv  = __shfl(g[r & 7], (r < 8) ? 0 : 16, 32) * mk;
        const float4 xv = *(const float4*)(tile + r * kDim + lane * 4);
        if (seg != curSeg) {
          float* p = out + (size_t)curSeg * kDim + lane * 4;
          unsafeAtomicAdd(p + 0, ax); unsafeAtomicAdd(p + 1, ay);
          unsafeAtomicAdd(p + 2, az); unsafeAtomicAdd(p + 3, aw);
          ax = ay = az = aw = 0.f;
          curSeg = seg;
        }
        ax += gv * xv.x; ay += gv * xv.y; az += gv * xv.z; aw += gv * xv.w;
      }
    }
  }
  // final flush of the carried run
  float* p = out + (size_t)curSeg * kDim + lane * 4;
  unsafeAtomicAdd(p + 0, ax); unsafeAtomicAdd(p + 1, ay);
  unsafeAtomicAdd(p + 2, az); unsafeAtomicAdd(p + 3, aw);
}

__global__ void zero_kernel(float* p, int n) {
  int i = blockIdx.x * blockDim.x + threadIdx.x;
  if (i < n) p[i] = 0.0f;
}

extern "C" void kernel_launch(void* const* d_in, const int* in_sizes, int n_in,
                              void* d_out, int out_size, void* d_ws, size_t ws_size,
                              hipStream_t stream) {
  const float* x     = (const float*)d_in[0];
  const int*   batch = (const int*)d_in[1];
  const float* mask  = (const float*)d_in[2];
  const float* W     = (const float*)d_in[3];
  const float* bvec  = (const float*)d_in[4];
  float*       out   = (float*)d_out;
  const int N = in_sizes[1];   // batch has N elements

  zero_kernel<<<(out_size + 255) / 256, 256, 0, stream>>>(out, out_size);

  const int totalWaves = (N + kRowsPerWave - 1) / kRowsPerWave;
  const int blocks     = (totalWaves + kWavesPerBlk - 1) / kWavesPerBlk;
  wsum_kernel<<<blocks, 32 * kWavesPerBlk, 0, stream>>>(x, batch, mask, W, bvec, out, N);
}